// STSemModule_15118284882369
// MI455X (gfx1250) — compile-verified
//
#include <hip/hip_runtime.h>
#include <hip/hip_bf16.h>
#include <stdint.h>

// MI455X / gfx1250, wave32. Heavy math through V_WMMA_F32_16X16X4_F32 (exact
// fp32, matrix pipe). One generic 64x64-tile GEMM kernel; eligible launches
// stream tiles with GLOBAL_LOAD_ASYNC_TO_LDS_B128 (ASYNCcnt, double-buffered
// LDS); guarded launches (K=100 pad, TCN shifted gather) use a register-staged
// double-buffered fallback.

typedef float v2f __attribute__((ext_vector_type(2)));
typedef float v8f __attribute__((ext_vector_type(8)));
typedef int   v4i_vs __attribute__((vector_size(16)));   // builtin's pointee type

#define F_ACC  1
#define F_RELU 2

#define PA 20   // As pitch (floats): conflict-free 16-lane stride, 16B rows
#define PB 80   // Bs pitch (floats): +1 row = +16 banks, half-waves disjoint

#if defined(__has_builtin)
#  if __has_builtin(__builtin_amdgcn_global_load_async_to_lds_b128)
#    define ASY_BUILTIN 1
#  endif
#  if __has_builtin(__builtin_amdgcn_s_wait_asynccnt)
#    define WAITASY_BUILTIN 1
#  endif
#endif
#ifndef ASY_BUILTIN
#define ASY_BUILTIN 0
#endif
#ifndef WAITASY_BUILTIN
#define WAITASY_BUILTIN 0
#endif

__device__ __forceinline__ void async_cp16(const float* g, float* l)
{
#if ASY_BUILTIN
    __builtin_amdgcn_global_load_async_to_lds_b128(
        (__attribute__((address_space(1))) v4i_vs*)g,
        (__attribute__((address_space(3))) v4i_vs*)l, 0, 0);
#else
    // generic LDS pointer: low 32 bits are the LDS byte offset (ISA aperture)
    unsigned off = (unsigned)(uintptr_t)l;
    asm volatile("global_load_async_to_lds_b128 %0, %1, off"
                 :: "v"(off), "v"(g) : "memory");
#endif
}

__device__ __forceinline__ void wait_async_4()
{
#if WAITASY_BUILTIN
    __builtin_amdgcn_s_wait_asynccnt(4);
#else
    asm volatile("s_wait_asynccnt 0x4" ::: "memory");
#endif
}
__device__ __forceinline__ void wait_async_0()
{
#if WAITASY_BUILTIN
    __builtin_amdgcn_s_wait_asynccnt(0);
#else
    asm volatile("s_wait_asynccnt 0x0" ::: "memory");
#endif
}

__device__ __forceinline__ void compute_chunk(const float* AsB, const float* BsB,
                                              v8f* acc, int wv, int lr, int hk)
{
    #pragma unroll
    for (int kk = 0; kk < 4; ++kk) {
        const int k4 = kk * 4;
        v2f a;
        a.x = AsB[(wv * 16 + lr) * PA + k4 + hk];
        a.y = AsB[(wv * 16 + lr) * PA + k4 + hk + 1];
        #pragma unroll
        for (int s = 0; s < 4; ++s) {
            v2f b;
            b.x = BsB[(k4 + hk)     * PB + s * 16 + lr];
            b.y = BsB[(k4 + hk + 1) * PB + s * 16 + lr];
            acc[s] = __builtin_amdgcn_wmma_f32_16x16x4_f32(
                false, a, false, b, (short)0, acc[s], false, false);
        }
    }
}

__global__ __launch_bounds__(128)
void gemm64x64(const float* __restrict__ A, int lda,
               const float* __restrict__ B, int ldb,
               float* __restrict__ C, int ldc,
               int K,
               const float* __restrict__ bias,
               const float* __restrict__ res,
               int flags, int tcnD, int useAsync)
{
    // A: M x K (row major, lda)   [tcnD>0: gathered TCN weights, K=128]
    // B: K x N (row major, ldb)   [tcnD>0: rows >=64 are t-shifted by tcnD]
    // C: M x N (row major, ldc)
    __shared__ __align__(16) float As[2][64 * PA];
    __shared__ __align__(16) float Bs[2][16 * PB];

    const int tid  = threadIdx.x;
    const int lane = tid & 31;
    const int wv   = tid >> 5;          // wave 0..3 -> rows 16*wv..16*wv+15
    const int lr   = lane & 15;
    const int hk   = (lane >> 4) << 1;  // 0 or 2 (K-pair per half-wave)

    const long mBase = (long)blockIdx.y * 64;
    const long nBase = (long)blockIdx.x * 64;

    v8f acc[4] = {};
    const int kChunks = (K + 15) >> 4;

    // fallback staging indices
    const int ar  = tid >> 1;        // A: 2 threads/row, 8 k each
    const int akb = (tid & 1) * 8;
    const int bkr = tid >> 3;        // B: 8 threads/row, 8 cols each
    const int bcb = (tid & 7) * 8;

#define STAGE_ASYNC(kc, buf)                                                  \
  {                                                                           \
    const int k0a = (kc) << 4;                                                \
    _Pragma("unroll") for (int i = 0; i < 2; ++i) {                           \
      const int cid = tid + i * 128;                                          \
      const int row = cid >> 2;                                               \
      const int kq  = (cid & 3) << 2;                                         \
      async_cp16(A + (mBase + row) * (long)lda + k0a + kq,                    \
                 &As[buf][row * PA + kq]);                                    \
    }                                                                         \
    _Pragma("unroll") for (int i = 0; i < 2; ++i) {                           \
      const int cid = tid + i * 128;                                          \
      const int kr  = cid >> 4;                                               \
      const int cq  = (cid & 15) << 2;                                        \
      async_cp16(B + (long)(k0a + kr) * ldb + nBase + cq,                     \
                 &Bs[buf][kr * PB + cq]);                                     \
    }                                                                         \
  }

#define LOAD_REGS(kc)                                                         \
  {                                                                           \
    const int k0a = (kc) << 4;                                                \
    _Pragma("unroll") for (int j = 0; j < 8; ++j) {                           \
      const int gk = k0a + akb + j;                                           \
      float v = 0.0f;                                                         \
      if (gk < K) {                                                           \
        if (tcnD == 0) {                                                      \
          v = A[(mBase + ar) * (long)lda + gk];                               \
        } else {                                                              \
          const int i6 = gk & 63;                                             \
          v = A[(((long)ar * 64 + i6) << 1) + ((gk < 64) ? 1 : 0)];           \
        }                                                                     \
      }                                                                       \
      ra[j] = v;                                                              \
    }                                                                         \
    _Pragma("unroll") for (int j = 0; j < 8; ++j) {                           \
      const long gc = nBase + bcb + j;                                        \
      const int gk = k0a + bkr;                                               \
      float v = 0.0f;                                                         \
      if (gk < K) {                                                           \
        if (tcnD == 0 || gk < 64) {                                           \
          v = B[(long)gk * ldb + gc];                                         \
        } else {                                                              \
          const int tt = (int)((gc >> 10) % 12);                              \
          if (tt >= tcnD)                                                     \
            v = B[(long)(gk - 64) * ldb + gc - ((long)tcnD << 10)];           \
        }                                                                     \
      }                                                                       \
      rb[j] = v;                                                              \
    }                                                                         \
  }

#define STORE_REGS(buf)                                                       \
  {                                                                           \
    _Pragma("unroll") for (int j = 0; j < 8; ++j)                             \
      As[buf][ar * PA + akb + j] = ra[j];                                     \
    _Pragma("unroll") for (int j = 0; j < 8; ++j)                             \
      Bs[buf][bkr * PB + bcb + j] = rb[j];                                    \
  }

    if (useAsync) {
        // DMA path: GLOBAL_LOAD_ASYNC_TO_LDS_B128, double-buffered LDS.
        STAGE_ASYNC(0, 0);
        for (int kc = 0; kc < kChunks; ++kc) {
            const int cur = kc & 1;
            if (kc + 1 < kChunks) {
                STAGE_ASYNC(kc + 1, cur ^ 1);
                wait_async_4();          // first 4 (current buffer) done
            } else {
                wait_async_0();
            }
            __syncthreads();
            compute_chunk(&As[cur][0], &Bs[cur][0], acc, wv, lr, hk);
            __syncthreads();
        }
    } else {
        // guarded fallback: register-staged, double-buffered
        float ra[8], rb[8];
        LOAD_REGS(0);
        STORE_REGS(0);
        for (int kc = 0; kc < kChunks; ++kc) {
            const int cur = kc & 1;
            if (kc + 1 < kChunks) LOAD_REGS(kc + 1);
            __syncthreads();
            compute_chunk(&As[cur][0], &Bs[cur][0], acc, wv, lr, hk);
            if (kc + 1 < kChunks) STORE_REGS(cur ^ 1);
        }
        __syncthreads();
    }

    // ---- epilogue: bias / accumulate / relu / residual-add-relu ----
    const int rofs = (lane >> 4) << 3;   // 0 or 8
    #pragma unroll
    for (int s = 0; s < 4; ++s) {
        const long col = nBase + s * 16 + lr;
        #pragma unroll
        for (int i = 0; i < 8; ++i) {
            const long row = mBase + wv * 16 + i + rofs;
            float v = acc[s][i];
            if (bias)           v += bias[row];
            if (flags & F_ACC)  v += C[row * (long)ldc + col];
            if (flags & F_RELU) v  = fmaxf(v, 0.0f);
            if (res)            v  = fmaxf(v + res[row * (long)ldc + col], 0.0f);
            C[row * (long)ldc + col] = v;
        }
    }
#undef STAGE_ASYNC
#undef LOAD_REGS
#undef STORE_REGS
}

// row-wise softmax over 1024 columns (values already relu'd >= 0)
__global__ void softmax_row(float* __restrict__ data, int ncols)
{
    __shared__ float red[256];
    float* p = data + (size_t)blockIdx.x * ncols;
    const int tid = threadIdx.x;
    float m = -1e30f;
    for (int i = tid; i < ncols; i += 256) m = fmaxf(m, p[i]);
    red[tid] = m; __syncthreads();
    for (int s = 128; s > 0; s >>= 1) {
        if (tid < s) red[tid] = fmaxf(red[tid], red[tid + s]);
        __syncthreads();
    }
    m = red[0]; __syncthreads();
    float sum = 0.0f;
    for (int i = tid; i < ncols; i += 256) sum += expf(p[i] - m);
    red[tid] = sum; __syncthreads();
    for (int s = 128; s > 0; s >>= 1) {
        if (tid < s) red[tid] += red[tid + s];
        __syncthreads();
    }
    const float inv = 1.0f / red[0];
    __syncthreads();
    for (int i = tid; i < ncols; i += 256) p[i] = expf(p[i] - m) * inv;
}

// (bt, d, n) -> (d, bt, n): x0w[d*98304 + bt*1024 + n] = gf[(bt*48+d)*1024 + n]
__global__ void transpose_in(const float* __restrict__ gf, float* __restrict__ xw, int nElem)
{
    int idx = blockIdx.x * blockDim.x + threadIdx.x;
    if (idx >= nElem) return;
    int d  = idx / (96 * 1024);
    int r  = idx - d * (96 * 1024);
    int bt = r >> 10;
    int n  = r & 1023;
    xw[idx] = gf[((bt * 48 + d) << 10) + n];
}

// GT[n][w] = G[w][n] (1024 x 1024)
__global__ void transpose_sq(const float* __restrict__ G, float* __restrict__ GT)
{
    int idx = blockIdx.x * blockDim.x + threadIdx.x;
    int n = idx >> 10, w = idx & 1023;
    GT[idx] = G[(w << 10) + n];
}

// lastT[(b*64+e)][n] = tcn[e][(b*12+11)*1024 + n]
__global__ void gather_last(const float* __restrict__ tcn, float* __restrict__ lastT)
{
    int idx = blockIdx.x * blockDim.x + threadIdx.x;
    if (idx >= 512 * 1024) return;
    int row = idx >> 10;
    int n   = idx & 1023;
    int b = row >> 6, e = row & 63;
    lastT[idx] = tcn[(size_t)e * 98304 + (size_t)((b * 12 + 11) << 10) + n];
}

extern "C" void kernel_launch(void* const* d_in, const int* in_sizes, int n_in,
                              void* d_out, int out_size, void* d_ws, size_t ws_size,
                              hipStream_t stream)
{
    const float* gf  = (const float*)d_in[0];  // (8,12,48,1024)
    const float* G   = (const float*)d_in[1];  // (1024,1024)
    const float* nv1 = (const float*)d_in[2];  // (3,1024,100)
    const float* nv2 = (const float*)d_in[3];  // (3,100,1024)
    const float* w0  = (const float*)d_in[4];  // (3,64,144)
    const float* b0  = (const float*)d_in[5];  // (3,64)
    const float* w1  = (const float*)d_in[6];  // (3,64,192)
    const float* b1  = (const float*)d_in[7];  // (3,64)
    const float* tw  = (const float*)d_in[8];  // (3,2,64,64,2)
    const float* tb  = (const float*)d_in[9];  // (3,2,64)
    float* out = (float*)d_out;

    // workspace layout (floats); total 33,030,144 floats = ~132 MB
    if (ws_size < 33030144ull * sizeof(float)) return;
    float* ws    = (float*)d_ws;
    float* adp   = ws;                   // 3 * 1048576
    float* x0w   = adp + 3 * 1048576;    // 48 * 98304 = 4718592
    float* x1buf = x0w + 4718592;        // 6144 * 1024
    float* x2buf = x1buf + 6291456;      // 6144 * 1024
    float* out0  = x2buf + 6291456;      // 64 * 98304
    float* out1  = out0 + 6291456;       // 64 * 98304

    dim3 blk(128);

    // ---- 1) adaptive adjacencies: relu(nv1 @ nv2) then row softmax ----
    // K=100 (padded) -> fallback path
    for (int s = 0; s < 3; ++s) {
        gemm64x64<<<dim3(16, 16), blk, 0, stream>>>(
            nv1 + (size_t)s * 1024 * 100, 100,
            nv2 + (size_t)s * 100 * 1024, 1024,
            adp + (size_t)s * 1048576, 1024,
            100, nullptr, nullptr, F_RELU, 0, 0);
    }
    softmax_row<<<3 * 1024, 256, 0, stream>>>(adp, 1024);

    // ---- 2) input to channel-major (d, bt, n) ----
    transpose_in<<<(4718592 + 255) / 256, 256, 0, stream>>>(gf, x0w, 4718592);

    // ---- 3) GCN layer 0 (C=48) ----
    for (int s = 0; s < 3; ++s) {
        const float* A = adp + (size_t)s * 1048576;
        gemm64x64<<<dim3(16, 72), blk, 0, stream>>>(x0w,  1024, A, 1024, x1buf, 1024, 1024, nullptr, nullptr, 0, 0, 1);
        gemm64x64<<<dim3(16, 72), blk, 0, stream>>>(x1buf,1024, A, 1024, x2buf, 1024, 1024, nullptr, nullptr, 0, 0, 1);
        gemm64x64<<<dim3(1536, 1), blk, 0, stream>>>(w0 + (size_t)s*64*144 +  0, 144, x0w,   98304, out0, 98304, 48, b0 + s*64, nullptr, (s ? F_ACC : 0), 0, 1);
        gemm64x64<<<dim3(1536, 1), blk, 0, stream>>>(w0 + (size_t)s*64*144 + 48, 144, x1buf, 98304, out0, 98304, 48, nullptr, nullptr, F_ACC, 0, 1);
        gemm64x64<<<dim3(1536, 1), blk, 0, stream>>>(w0 + (size_t)s*64*144 + 96, 144, x2buf, 98304, out0, 98304, 48, nullptr, nullptr, F_ACC, 0, 1);
    }
    // ---- GCN layer 1 (C=64) ----
    for (int s = 0; s < 3; ++s) {
        const float* A = adp + (size_t)s * 1048576;
        gemm64x64<<<dim3(16, 96), blk, 0, stream>>>(out0, 1024, A, 1024, x1buf, 1024, 1024, nullptr, nullptr, 0, 0, 1);
        gemm64x64<<<dim3(16, 96), blk, 0, stream>>>(x1buf,1024, A, 1024, x2buf, 1024, 1024, nullptr, nullptr, 0, 0, 1);
        gemm64x64<<<dim3(1536, 1), blk, 0, stream>>>(w1 + (size_t)s*64*192 +   0, 192, out0,  98304, out1, 98304, 64, b1 + s*64, nullptr, (s ? F_ACC : 0), 0, 1);
        gemm64x64<<<dim3(1536, 1), blk, 0, stream>>>(w1 + (size_t)s*64*192 +  64, 192, x1buf, 98304, out1, 98304, 64, nullptr, nullptr, F_ACC, 0, 1);
        gemm64x64<<<dim3(1536, 1), blk, 0, stream>>>(w1 + (size_t)s*64*192 + 128, 192, x2buf, 98304, out1, 98304, 64, nullptr, nullptr, F_ACC, 0, 1);
    }

    // ---- 4) TCN: 3 levels, dilation 1,2,4 (shifted gather -> fallback) ----
    {
        float* inB[3]  = { out1,  x2buf, x1buf };
        float* h1B[3]  = { x1buf, out1,  x2buf };
        float* h2B[3]  = { x2buf, x1buf, out1  };   // final result lands in out1
        for (int lvl = 0; lvl < 3; ++lvl) {
            const int d = 1 << lvl;
            const float* W1 = tw + (size_t)(lvl * 2 + 0) * 64 * 64 * 2;
            const float* W2 = tw + (size_t)(lvl * 2 + 1) * 64 * 64 * 2;
            const float* Bb1 = tb + (lvl * 2 + 0) * 64;
            const float* Bb2 = tb + (lvl * 2 + 1) * 64;
            gemm64x64<<<dim3(1536, 1), blk, 0, stream>>>(W1, 0, inB[lvl], 98304, h1B[lvl], 98304, 128, Bb1, nullptr,  F_RELU, d, 0);
            gemm64x64<<<dim3(1536, 1), blk, 0, stream>>>(W2, 0, h1B[lvl], 98304, h2B[lvl], 98304, 128, Bb2, inB[lvl], F_RELU, d, 0);
        }
    }

    // ---- 5) last timestep -> grid map GEMM -> d_out (b,e,w) ----
    float* lastT = x0w;   // 512 x 1024 (region dead)
    float* GT    = out0;  // 1024 x 1024 (region dead)
    gather_last<<<(512 * 1024) / 256, 256, 0, stream>>>(out1, lastT);
    transpose_sq<<<(1024 * 1024) / 256, 256, 0, stream>>>(G, GT);
    gemm64x64<<<dim3(16, 8), blk, 0, stream>>>(lastT, 1024, GT, 1024, out, 1024, 1024, nullptr, nullptr, 0, 0, 1);
}